// SSIMLoss_39668317946108
// MI455X (gfx1250) — compile-verified
//
#include <hip/hip_runtime.h>
#include <hip/hip_bf16.h>
#include <math.h>

// SSIM loss for [16,3,512,512] f32 via separable 11x11 Gaussian blur,
// implemented as banded-matrix WMMA f32 16x16x4 matmuls (CDNA5 / gfx1250).
// Each wave produces a 16x32 output region (2 output tiles) from a shared
// 26x42 halo patch: pass1 (vertical) = 3 N-tiles x 7 K-chunks, pass2
// (horizontal) = 2 tiles x 7 K-chunks -> 35 WMMA per map, 175 per wave.
// Patches are stored pair-interleaved in K so every WMMA B fragment is one
// 8-byte-aligned ds_load_b64 (no operand shuffle movs).

typedef float v2f __attribute__((ext_vector_type(2)));
typedef float v8f __attribute__((ext_vector_type(8)));

#define SSIM_C1 1e-4f     // 0.01^2
#define SSIM_C2 9e-4f     // 0.03^2

#define IMG_H   512
#define IMG_W   512
#define NPLANES 48                       // 16 batch * 3 channels
#define BLK_PER_PLANE 512                // 32 row-tiles x 16 col-groups (16x32 px)
#define NBLOCKS (NPLANES * BLK_PER_PLANE)   // 24576
#define NPIXELS 12582912.0f              // 16*3*512*512

// Pair-interleaved patch layout: word addr = (j>>1)*PP_STRIDE + c*2 + (j&1)
// PP_STRIDE = 96 -> lanes16-31 (next pair-row) sit +96 = +32 banks: conflict-free.
#define PP_STRIDE 96                     // 2 * 48 cols
#define PP_PAIRS  14                     // 28 K rows
#define P_WORDS   (PP_PAIRS * PP_STRIDE) // 1344 (multiple of 64)
// V_STRIDE = 52: 52*L mod 64 distinct for L=0..15; EXEC halves disjoint mod 4;
// wide enough for all 48 columns so tile-2 stores need no clipping/branches.
#define V_STRIDE  52
#define RAW_W     42                     // patch cols = 32 + 10 halo
#define RAW_H     26                     // patch rows = 16 + 10 halo

__global__ __launch_bounds__(32)
void ssim_tile_kernel(const float* __restrict__ S,
                      const float* __restrict__ T,
                      const float* __restrict__ K2D,
                      float* __restrict__ partial)
{
    __shared__ float g[16];              // normalized 1D gaussian
    __shared__ float Ps[P_WORDS];        // s patch, K-pair-interleaved, zero padded
    __shared__ float Pt[P_WORDS];        // t patch
    __shared__ float Pm[P_WORDS];        // product map (ss/tt/st)
    __shared__ float Vb[16 * V_STRIDE];  // vertical-blur intermediate (row-major)
    __shared__ float red[32];

    const int lane = threadIdx.x;
    const int job  = blockIdx.x;             // one 16x32 region per wave
    const int plane = job >> 9;              // / 512
    const int rem   = job & 511;
    const int tR = rem >> 4, tP = rem & 15;
    const int planeBase = plane * (IMG_H * IMG_W);
    const int R0 = tR * 16 - 5, C0 = tP * 32 - 5;   // patch origin (halo 5)

    // Recover normalized 1D kernel: k2d[i][i] = (g[i]/sum(g))^2
    if (lane < 11) g[lane] = sqrtf(K2D[lane * 12]);

    // Zero Ps/Pt in b64 chunks (pads must stay zero; 1344/64 = 21 iterations).
    {
        v2f z = {0.0f, 0.0f};
        for (int e = lane * 2; e < P_WORDS; e += 64) {
            *(v2f*)(Ps + e) = z;
            *(v2f*)(Pt + e) = z;
        }
    }

    // Load raw 26x42 patches into pair-interleaved layout with zero halo.
    for (int e = lane; e < RAW_H * RAW_W; e += 32) {
        int pr = e / RAW_W, pc = e - pr * RAW_W;
        int gr = R0 + pr, gc = C0 + pc;
        bool in = ((unsigned)gr < (unsigned)IMG_H) && ((unsigned)gc < (unsigned)IMG_W);
        int idx = planeBase + gr * IMG_W + gc;
        int lp  = (pr >> 1) * PP_STRIDE + pc * 2 + (pr & 1);
        Ps[lp] = in ? S[idx] : 0.0f;
        Pt[lp] = in ? T[idx] : 0.0f;
    }
    __syncthreads();

    // Per-lane band fragments, shared by pass-1 A and pass-2 B (and all tiles):
    // value = a[(4*kc + koff + {0,1}) - (lane&15)] within band [0,10].
    // A layout (16x4 f32, documented): lane<16 holds {K=j0,j0+1} of row lane,
    // lane>=16 holds {j0+2,j0+3}; B layout assumed mirrored.
    const int rsel = lane & 15;
    const int koff = (lane >= 16) ? 2 : 0;
    float a0[7], a1[7];
#pragma unroll
    for (int kc = 0; kc < 7; ++kc) {
        int d0 = 4 * kc + koff - rsel;
        a0[kc] = (d0 >= 0 && d0 <= 10) ? g[d0] : 0.0f;
        int d1 = d0 + 1;
        a1[kc] = (d1 >= 0 && d1 <= 10) ? g[d1] : 0.0f;
    }

    v8f acc[5][2];   // [map][out tile]

#pragma unroll
    for (int m = 0; m < 5; ++m) {
        // Select/build this map's patch (interleaving is transparent to
        // elementwise products; pads are zero so products of pads stay zero).
        const float* Pcur;
        if (m == 0)      Pcur = Ps;
        else if (m == 1) Pcur = Pt;
        else {
            const float* X = (m == 3) ? Pt : Ps;     // ss:Ps*Ps tt:Pt*Pt st:Ps*Pt
            const float* Y = (m == 2) ? Ps : Pt;
            for (int e = lane * 2; e < P_WORDS; e += 64) {
                v2f x = *(const v2f*)(X + e);
                v2f y = *(const v2f*)(Y + e);
                v2f p; p.x = x.x * y.x; p.y = x.y * y.y;
                *(v2f*)(Pm + e) = p;
            }
            Pcur = Pm;
            __syncthreads();
        }

        // Pass 1 (vertical): V[r,c] = sum_j a[j-r] * P[j,c], 3 N-tiles of 16.
        // B fragment {P[j0+koff,c], P[j0+koff+1,c]} = one contiguous b64.
        v8f v0 = {0, 0, 0, 0, 0, 0, 0, 0};
        v8f v1 = v0, v2 = v0;
#pragma unroll
        for (int kc = 0; kc < 7; ++kc) {
            v2f A; A.x = a0[kc]; A.y = a1[kc];
            int base = (2 * kc + (koff >> 1)) * PP_STRIDE + rsel * 2;
            v2f B0 = *(const v2f*)(Pcur + base);
            v2f B1 = *(const v2f*)(Pcur + base + 32);
            v2f B2 = *(const v2f*)(Pcur + base + 64);
            v0 = __builtin_amdgcn_wmma_f32_16x16x4_f32(false, A, false, B0, (short)0, v0, false, false);
            v1 = __builtin_amdgcn_wmma_f32_16x16x4_f32(false, A, false, B1, (short)0, v1, false, false);
            v2 = __builtin_amdgcn_wmma_f32_16x16x4_f32(false, A, false, B2, (short)0, v2, false, false);
        }
        // Store V via documented C/D layout: VGPR i -> row i (lanes 0-15) / i+8.
        // Stride 52 fits all 48 cols, so stores are uniform (no EXEC clipping);
        // cols 42..47 are exact zeros (patch pad) and never used with nonzero
        // band coefficients.
        {
            int row0 = (lane >= 16) ? 8 : 0;
#pragma unroll
            for (int i = 0; i < 8; ++i) {
                int ra = (row0 + i) * V_STRIDE + rsel;
                Vb[ra]      = v0[i];
                Vb[ra + 16] = v1[i];
                Vb[ra + 32] = v2[i];
            }
        }
        __syncthreads();

        // Pass 2 (horizontal): O[r, 16*ot + n] = sum_c V[r, 16*ot + c] * a[c - n]
        // A fragment {V[r,cb], V[r,cb+1]} = one contiguous b64 (cb even).
#pragma unroll
        for (int ot = 0; ot < 2; ++ot) {
            v8f o = {0, 0, 0, 0, 0, 0, 0, 0};
#pragma unroll
            for (int kc = 0; kc < 7; ++kc) {
                int cb = rsel * V_STRIDE + 16 * ot + 4 * kc + koff;
                v2f A2 = *(const v2f*)(Vb + cb);
                v2f B2; B2.x = a0[kc]; B2.y = a1[kc];
                o = __builtin_amdgcn_wmma_f32_16x16x4_f32(false, A2, false, B2, (short)0, o, false, false);
            }
            acc[m][ot] = o;
        }
        __syncthreads();   // Vb (and Pm) are rewritten by the next map
    }

    // SSIM epilogue on the 16 (r,c) elements this lane owns.
    float sum = 0.0f;
#pragma unroll
    for (int ot = 0; ot < 2; ++ot) {
#pragma unroll
        for (int i = 0; i < 8; ++i) {
            float ux = acc[0][ot][i], uy = acc[1][ot][i];
            float uxx = acc[2][ot][i], uyy = acc[3][ot][i], uxy = acc[4][ot][i];
            float uxuy = ux * uy;
            float num = (2.0f * uxuy + SSIM_C1) * (2.0f * (uxy - uxuy) + SSIM_C2);
            float den = (ux * ux + uy * uy + SSIM_C1)
                      * ((uxx - ux * ux) + (uyy - uy * uy) + SSIM_C2);
            sum += num / den;
        }
    }
    red[lane] = sum;
    __syncthreads();
    if (lane == 0) {
        float t = 0.0f;
        for (int i = 0; i < 32; ++i) t += red[i];   // fixed order: deterministic
        partial[job] = t;
    }
}

__global__ __launch_bounds__(256)
void ssim_reduce_kernel(const float* __restrict__ partial, float* __restrict__ out, int n)
{
    __shared__ float red[256];
    float s = 0.0f;
    for (int i = threadIdx.x; i < n; i += 256) s += partial[i];   // fixed order
    red[threadIdx.x] = s;
    __syncthreads();
    for (int off = 128; off > 0; off >>= 1) {
        if ((int)threadIdx.x < off) red[threadIdx.x] += red[threadIdx.x + off];
        __syncthreads();
    }
    if (threadIdx.x == 0) out[0] = 1.0f - red[0] / NPIXELS;
}

extern "C" void kernel_launch(void* const* d_in, const int* in_sizes, int n_in,
                              void* d_out, int out_size, void* d_ws, size_t ws_size,
                              hipStream_t stream) {
    const float* S   = (const float*)d_in[0];   // img_output [16,3,512,512]
    const float* T   = (const float*)d_in[1];   // img_target [16,3,512,512]
    const float* K2D = (const float*)d_in[2];   // kernel [3,1,11,11]
    float* out = (float*)d_out;
    float* partial = (float*)d_ws;              // NBLOCKS floats = 96 KB

    ssim_tile_kernel<<<NBLOCKS, 32, 0, stream>>>(S, T, K2D, partial);
    ssim_reduce_kernel<<<1, 256, 0, stream>>>(partial, out, NBLOCKS);
}